// Signature_50543175139965
// MI455X (gfx1250) — compile-verified
//
#include <hip/hip_runtime.h>

typedef __attribute__((ext_vector_type(2))) float v2f;
typedef __attribute__((ext_vector_type(8))) float v8f;

#define TPAD 256   // 255 real increments, padded with dx=0
#define NC 8
#define OUT_PER 4680
#define OFF_S1 0
#define OFF_S2 8
#define OFF_S3 72
#define OFF_S4 584

// Branchless A-fragment element for k-index t (compile-time PHASE).
template <int PHASE>
__device__ __forceinline__ float aElem(int t, int m, float selA, float addA,
                                       const float* __restrict__ dxs,
                                       const float* __restrict__ p1m,
                                       const float* __restrict__ p2m) {
  if (PHASE == 0 || PHASE == 1 || PHASE == 5) {
    // rows 0-7 = P1m, (phases 0/1) row 8 = 1, rest 0 -- via lane masks
    return selA * p1m[t * NC + (m & 7)] + addA;
  } else if (PHASE == 2 || PHASE == 4) {
    return p2m[t * 64 + m];                       // m < 64 always
  } else {  // PHASE 3: e3 on the fly
    const int a = (m >> 6) & 7, b = (m >> 3) & 7, c = m & 7;
    return dxs[t * NC + a] * dxs[t * NC + b] * dxs[t * NC + c] * (1.0f / 6.0f);
  }
}

// Branchless B-fragment element for k-index t (compile-time PHASE).
template <int PHASE>
__device__ __forceinline__ float bElem(int t, int n, float selB, float addB,
                                       const float* __restrict__ dxs,
                                       const float* __restrict__ r1s) {
  if (PHASE == 0 || PHASE == 2) {
    return selB * dxs[t * NC + (n & 7)];
  } else if (PHASE == 1) {
    const int b = (n >> 3) & 7, c = n & 7;
    return 0.5f * dxs[t * NC + b] * dxs[t * NC + c];
  } else if (PHASE == 3) {
    const int d = n & 7;
    return selB * (r1s[t * NC + d] + 0.25f * dxs[t * NC + d]) + addB;
  } else if (PHASE == 4) {
    const int c = (n >> 3) & 7, d = n & 7;
    return dxs[t * NC + c] * (r1s[t * NC + d] + 0.5f * dxs[t * NC + d]);
  } else {  // PHASE 5
    const int b = (n >> 6) & 7, c = (n >> 3) & 7, d = n & 7;
    return dxs[t * NC + b] * dxs[t * NC + c] *
           (0.5f * r1s[t * NC + d] + (1.0f / 6.0f) * dxs[t * NC + d]);
  }
}

template <int PHASE>
__device__ __forceinline__ void doTile(int mt, int nt, int lane,
                                       const float* __restrict__ dxs,
                                       const float* __restrict__ p1m,
                                       const float* __restrict__ r1s,
                                       const float* __restrict__ p2m,
                                       float* __restrict__ sacc) {
  const int hh = lane >> 4;   // half-wave: K sub-pair select (A/B), M+8 (D)
  const int q  = lane & 15;
  const int m  = mt * 16 + q; // A row
  const int n  = nt * 16 + q; // B column

  // Lane-constant masks (hoisted out of K loop; no EXEC branching inside).
  const float selA = (q < 8) ? 1.0f : 0.0f;                     // phases 0,1,5
  const float addA = (PHASE <= 1 && q == 8) ? 1.0f : 0.0f;      // ones row
  const float selB = (q < 8) ? 1.0f : 0.0f;                     // phases 0,2,3
  const float addB = (PHASE == 3 && q == 8) ? 1.0f : 0.0f;      // ones column

  v8f acc0 = {0.f, 0.f, 0.f, 0.f, 0.f, 0.f, 0.f, 0.f};
  v8f acc1 = {0.f, 0.f, 0.f, 0.f, 0.f, 0.f, 0.f, 0.f};

#pragma unroll 4
  for (int k0 = 0; k0 < TPAD; k0 += 8) {
    {
      const int ka = k0 + hh * 2;  // vgpr j holds K = j + 2*(lane>=16)
      v2f av, bv;
      av.x = aElem<PHASE>(ka,     m, selA, addA, dxs, p1m, p2m);
      av.y = aElem<PHASE>(ka + 1, m, selA, addA, dxs, p1m, p2m);
      bv.x = bElem<PHASE>(ka,     n, selB, addB, dxs, r1s);
      bv.y = bElem<PHASE>(ka + 1, n, selB, addB, dxs, r1s);
      acc0 = __builtin_amdgcn_wmma_f32_16x16x4_f32(
          false, av, false, bv, (short)0, acc0, false, false);
    }
    {
      const int kb = k0 + 4 + hh * 2;
      v2f av, bv;
      av.x = aElem<PHASE>(kb,     m, selA, addA, dxs, p1m, p2m);
      av.y = aElem<PHASE>(kb + 1, m, selA, addA, dxs, p1m, p2m);
      bv.x = bElem<PHASE>(kb,     n, selB, addB, dxs, r1s);
      bv.y = bElem<PHASE>(kb + 1, n, selB, addB, dxs, r1s);
      acc1 = __builtin_amdgcn_wmma_f32_16x16x4_f32(
          false, av, false, bv, (short)0, acc1, false, false);
    }
  }

  // Writeback (cold path): D vgpr r, lane -> (M = r + 8*hh, N = q).
#pragma unroll
  for (int r = 0; r < 8; ++r) {
    const int mrow = r + 8 * hh;
    const float v = acc0[r] + acc1[r];
    if (PHASE == 0) {
      if (mrow < 8 && q < 8)       atomicAdd(&sacc[OFF_S2 + mrow * 8 + q], v);
      else if (mrow == 8 && q < 8) atomicAdd(&sacc[OFF_S1 + q], v);
    } else if (PHASE == 1) {
      const int gn = nt * 16 + q;
      if (mrow < 8)       atomicAdd(&sacc[OFF_S3 + mrow * 64 + gn], v);
      else if (mrow == 8) atomicAdd(&sacc[OFF_S2 + gn], v);
    } else if (PHASE == 2) {
      if (q < 8) atomicAdd(&sacc[OFF_S3 + (mt * 16 + mrow) * 8 + q], v);
    } else if (PHASE == 3) {
      const int gm = mt * 16 + mrow;
      if (q < 8)       atomicAdd(&sacc[OFF_S4 + gm * 8 + q], v);
      else if (q == 8) atomicAdd(&sacc[OFF_S3 + gm], v);
    } else if (PHASE == 4) {
      atomicAdd(&sacc[OFF_S4 + (mt * 16 + mrow) * 64 + nt * 16 + q], v);
    } else {
      if (mrow < 8) atomicAdd(&sacc[OFF_S4 + mrow * 512 + nt * 16 + q], v);
    }
  }
}

__global__ __launch_bounds__(256) void sig4_wmma_kernel(
    const float* __restrict__ path, float* __restrict__ out) {
  __shared__ float dxs[TPAD * NC];    //  8 KB  segment increments (t=255 zero)
  __shared__ float p1m[TPAD * NC];    //  8 KB  exclusive prefix of dx
  __shared__ float r1s[TPAD * NC];    //  8 KB  exclusive suffix of dx
  __shared__ float p2m[TPAD * 64];    // 64 KB  exclusive prefix of level-2
  __shared__ float sacc[OUT_PER];     // ~19 KB signature accumulator

  const int n = blockIdx.x;
  const int tid = threadIdx.x;
  const int lane = tid & 31;
  const int wave = tid >> 5;
  const float* p = path + (size_t)n * 256 * NC;

  // --- stage 1: increments + zero accumulator ---
  for (int i = tid; i < TPAD * NC; i += 256) {
    int t = i >> 3, c = i & 7;
    dxs[i] = (t < 255) ? (p[(t + 1) * NC + c] - p[t * NC + c]) : 0.0f;
  }
  for (int i = tid; i < OUT_PER; i += 256) sacc[i] = 0.0f;
  __syncthreads();

  // --- stage 2: exclusive prefix (P1m) and exclusive suffix (R1s) per channel ---
  if (tid < NC) {
    const int c = tid;
    float acc = 0.0f;
    for (int t = 0; t < TPAD; ++t) { p1m[t * NC + c] = acc; acc += dxs[t * NC + c]; }
  } else if (tid < 2 * NC) {
    const int c = tid - NC;
    float acc = 0.0f;
    for (int t = TPAD - 1; t >= 0; --t) { r1s[t * NC + c] = acc; acc += dxs[t * NC + c]; }
  }
  __syncthreads();

  // --- stage 3: exclusive prefix of level-2 (64 independent cumsums) ---
  if (tid < 64) {
    const int a = tid >> 3, b = tid & 7;
    float acc = 0.0f;
    for (int t = 0; t < TPAD; ++t) {
      p2m[t * 64 + tid] = acc;
      float da = dxs[t * NC + a], db = dxs[t * NC + b];
      acc += p1m[t * NC + a] * db + 0.5f * da * db;
    }
  }
  __syncthreads();

  // --- stage 4: 89 WMMA tiles, waves strided; wave-uniform phase dispatch ---
  for (int idx = wave; idx < 89; idx += 8) {
    if (idx == 0) {
      doTile<0>(0, 0, lane, dxs, p1m, r1s, p2m, sacc);
    } else if (idx < 5) {
      doTile<1>(0, idx - 1, lane, dxs, p1m, r1s, p2m, sacc);
    } else if (idx < 9) {
      doTile<2>(idx - 5, 0, lane, dxs, p1m, r1s, p2m, sacc);
    } else if (idx < 41) {
      doTile<3>(idx - 9, 0, lane, dxs, p1m, r1s, p2m, sacc);
    } else if (idx < 57) {
      doTile<4>((idx - 41) >> 2, (idx - 41) & 3, lane, dxs, p1m, r1s, p2m, sacc);
    } else {
      doTile<5>(0, idx - 57, lane, dxs, p1m, r1s, p2m, sacc);
    }
  }
  __syncthreads();

  // --- stage 5: write (S1|S2|S3|S4) for this batch ---
  float* o = out + (size_t)n * OUT_PER;
  for (int i = tid; i < OUT_PER; i += 256) o[i] = sacc[i];
}

extern "C" void kernel_launch(void* const* d_in, const int* in_sizes, int n_in,
                              void* d_out, int out_size, void* d_ws, size_t ws_size,
                              hipStream_t stream) {
  const float* path = (const float*)d_in[0];
  float* out = (float*)d_out;
  const int nb = in_sizes[0] / (256 * NC);  // N = 32
  sig4_wmma_kernel<<<nb, 256, 0, stream>>>(path, out);
}